// MDA_61306363183500
// MI455X (gfx1250) — compile-verified
//
#include <hip/hip_runtime.h>
#include <hip/hip_bf16.h>

// MI455X (gfx1250) implementation of the MDA block.
// Math simplifications (exact):
//   * all softmax-over-size-1 gates == 1.0
//   * H-branch == W-branch (same flat [B,8192,128] view since H==W) -> s = detal*(Co + 2*Wo)
// All GEMM stages use V_WMMA_F32_16X16X4_F32 (full fp32 precision, matrix pipe).
// Global->LDS staging uses GLOBAL_LOAD_ASYNC_TO_LDS_B128 (ASYNCcnt path).
// Workspace requirement: 17,104,896 floats = ~68.5 MB.

typedef float v2f __attribute__((ext_vector_type(2)));
typedef float v8f __attribute__((ext_vector_type(8)));

#define B_N   8
#define C_N   64
#define ICH   256
#define HW    16384
#define HWDIM 128

__device__ __forceinline__ v8f wmma4(v2f a, v2f b, v8f c) {
  // D = A(16x4,f32) * B(4x16,f32) + C(16x16,f32)
  return __builtin_amdgcn_wmma_f32_16x16x4_f32(false, a, false, b, (short)0, c,
                                               false, false);
}

__device__ __forceinline__ v8f vzero8() {
  v8f z = {0.f, 0.f, 0.f, 0.f, 0.f, 0.f, 0.f, 0.f};
  return z;
}

// Async direct global->LDS copy, 16 bytes per lane (both sides 16B aligned).
// Generic LDS pointers carry the LDS byte offset in their low 32 bits.
__device__ __forceinline__ void async_ld_b128(void* lds_dst, const void* gsrc) {
  uint32_t loff = (uint32_t)(uintptr_t)lds_dst;
  asm volatile("global_load_async_to_lds_b128 %0, %1, off"
               :: "v"(loff), "v"(gsrc)
               : "memory");
}
__device__ __forceinline__ void async_wait0() {
  asm volatile("s_wait_asynccnt 0" ::: "memory");
}

// ---------------------------------------------------------------------------
// Kernel 1: InConv  inp[b,m,p] = sum_i w_in[m,i]*x[b,i,p] + b_in[m]
// grid (128 pixel-tiles, 8 batches), block 256 (8 waves, 4 tiles each)
// ---------------------------------------------------------------------------
__global__ __launch_bounds__(256) void k_inconv(const float* __restrict__ x,
                                                const float* __restrict__ w_in,
                                                const float* __restrict__ b_in,
                                                float* __restrict__ inp) {
  __shared__ float xs[64][132];   // [k-chunk][pixel] (row 528B: 16B aligned, padded)
  __shared__ float wsm[64][68];   // [m][k-chunk]    (row 272B)
  const int b   = blockIdx.y;
  const int p0  = blockIdx.x * 128;
  const int tid = threadIdx.x, lane = tid & 31, wv = tid >> 5;
  const int hf  = lane >> 4, l16 = lane & 15;
  const int rb  = wv >> 1;          // M tile base = rb*16 (4 row-blocks)
  const int cb0 = (wv & 1) * 4;     // 4 consecutive N tiles
  v8f acc[4];
#pragma unroll
  for (int t = 0; t < 4; ++t) acc[t] = vzero8();

  for (int k0 = 0; k0 < ICH; k0 += 64) {
#pragma unroll
    for (int i = tid; i < 64 * 32; i += 256) {
      int kk = i >> 5, n4 = (i & 31) << 2;
      async_ld_b128(&xs[kk][n4],
                    x + ((size_t)(b * ICH + k0 + kk)) * HW + p0 + n4);
    }
#pragma unroll
    for (int i = tid; i < 64 * 16; i += 256) {
      int m = i >> 4, k4 = (i & 15) << 2;
      async_ld_b128(&wsm[m][k4], w_in + m * ICH + k0 + k4);
    }
    async_wait0();
    __syncthreads();
#pragma unroll
    for (int kk = 0; kk < 64; kk += 4) {
      v2f a;
      a.x = wsm[rb * 16 + l16][kk + 2 * hf];
      a.y = wsm[rb * 16 + l16][kk + 2 * hf + 1];
#pragma unroll
      for (int t = 0; t < 4; ++t) {
        const int nb = (cb0 + t) * 16;
        v2f bb;
        bb.x = xs[kk + 2 * hf][nb + l16];
        bb.y = xs[kk + 2 * hf + 1][nb + l16];
        acc[t] = wmma4(a, bb, acc[t]);
      }
    }
    __syncthreads();
  }
#pragma unroll
  for (int v = 0; v < 8; ++v) {
    const int m = rb * 16 + v + 8 * hf;
    const float bias = b_in[m];
#pragma unroll
    for (int t = 0; t < 4; ++t)
      inp[((size_t)(b * C_N + m)) * HW + p0 + (cb0 + t) * 16 + l16] = acc[t][v] + bias;
  }
}

// ---------------------------------------------------------------------------
// Kernel 2: GramW  G[b,w,v] += sum_k Xf[k,w]*Xf[k,v], Xf = inp[b] as [8192,128]
// grid (16 k-splits, 8 batches), block 1024 (32 waves, 2 of 64 tiles each)
// ---------------------------------------------------------------------------
__global__ __launch_bounds__(1024) void k_gramw(const float* __restrict__ inp,
                                                float* __restrict__ G) {
  __shared__ float xs[64][132];
  const int b   = blockIdx.y;
  const int ks  = blockIdx.x;
  const int tid = threadIdx.x, lane = tid & 31, wv = tid >> 5;
  const int hf  = lane >> 4, l16 = lane & 15;
  const int t0  = wv * 2;
  const int mb0 = (t0 >> 3) * 16,       nb0 = (t0 & 7) * 16;
  const int mb1 = ((t0 + 1) >> 3) * 16, nb1 = ((t0 + 1) & 7) * 16;
  v8f acc0 = vzero8(), acc1 = vzero8();

  const int kend = ks * 512 + 512;
  for (int k0 = ks * 512; k0 < kend; k0 += 64) {
#pragma unroll
    for (int i = tid; i < 64 * 32; i += 1024) {
      int kk = i >> 5, n4 = (i & 31) << 2;
      async_ld_b128(&xs[kk][n4],
                    inp + ((size_t)(b * 8192 + k0 + kk)) * HWDIM + n4);
    }
    async_wait0();
    __syncthreads();
#pragma unroll
    for (int kk = 0; kk < 64; kk += 4) {
      v2f a, bb;
      a.x  = xs[kk + 2 * hf][mb0 + l16]; a.y  = xs[kk + 2 * hf + 1][mb0 + l16];
      bb.x = xs[kk + 2 * hf][nb0 + l16]; bb.y = xs[kk + 2 * hf + 1][nb0 + l16];
      acc0 = wmma4(a, bb, acc0);
      a.x  = xs[kk + 2 * hf][mb1 + l16]; a.y  = xs[kk + 2 * hf + 1][mb1 + l16];
      bb.x = xs[kk + 2 * hf][nb1 + l16]; bb.y = xs[kk + 2 * hf + 1][nb1 + l16];
      acc1 = wmma4(a, bb, acc1);
    }
    __syncthreads();
  }
#pragma unroll
  for (int v = 0; v < 8; ++v) {
    atomicAdd(&G[((size_t)(b * 128 + mb0 + v + 8 * hf)) * 128 + nb0 + l16], acc0[v]);
    atomicAdd(&G[((size_t)(b * 128 + mb1 + v + 8 * hf)) * 128 + nb1 + l16], acc1[v]);
  }
}

// ---------------------------------------------------------------------------
// Kernel 3: GramC  G[b,c,d] += sum_p inp[b,c,p]*inp[b,d,p]
// grid (16 pixel-splits, 8 batches), block 512 (16 waves, 1 of 16 tiles each)
// ---------------------------------------------------------------------------
__global__ __launch_bounds__(512) void k_gramc(const float* __restrict__ inp,
                                               float* __restrict__ G) {
  __shared__ float ys[64][68];   // [channel][pixel-chunk]
  const int b   = blockIdx.y;
  const int ps  = blockIdx.x;
  const int tid = threadIdx.x, lane = tid & 31, wv = tid >> 5;
  const int hf  = lane >> 4, l16 = lane & 15;
  const int mb  = (wv >> 2) * 16, nb = (wv & 3) * 16;
  v8f acc = vzero8();

  const int pend = ps * 1024 + 1024;
  for (int p0 = ps * 1024; p0 < pend; p0 += 64) {
#pragma unroll
    for (int i = tid; i < 64 * 16; i += 512) {
      int e = i >> 4, n4 = (i & 15) << 2;
      async_ld_b128(&ys[e][n4], inp + ((size_t)(b * C_N + e)) * HW + p0 + n4);
    }
    async_wait0();
    __syncthreads();
#pragma unroll
    for (int kk = 0; kk < 64; kk += 4) {   // kk indexes pixels (the K dim)
      v2f a, bb;
      a.x  = ys[mb + l16][kk + 2 * hf]; a.y  = ys[mb + l16][kk + 2 * hf + 1];
      bb.x = ys[nb + l16][kk + 2 * hf]; bb.y = ys[nb + l16][kk + 2 * hf + 1];
      acc = wmma4(a, bb, acc);
    }
    __syncthreads();
  }
#pragma unroll
  for (int v = 0; v < 8; ++v)
    atomicAdd(&G[((size_t)(b * C_N + mb + v + 8 * hf)) * C_N + nb + l16], acc[v]);
}

// ---------------------------------------------------------------------------
// Kernel 4: column-wise stable softmax over axis=1 of [n,n]: T[r,v]=softmax_r
// grid (8 batches), block n threads (one thread per column v)
// ---------------------------------------------------------------------------
__global__ void k_colsoftmax(const float* __restrict__ G, float* __restrict__ T,
                             int n) {
  const int b = blockIdx.x, v = threadIdx.x;
  const float* g = G + (size_t)b * n * n;
  float* t = T + (size_t)b * n * n;
  float mx = -3.4e38f;
  for (int r = 0; r < n; ++r) mx = fmaxf(mx, g[r * n + v]);
  float s = 0.f;
  for (int r = 0; r < n; ++r) {
    float e = __expf(g[r * n + v] - mx);
    t[r * n + v] = e;
    s += e;
  }
  float inv = 1.f / s;
  for (int r = 0; r < n; ++r) t[r * n + v] *= inv;
}

__global__ void k_zero(float* __restrict__ p, int n) {
  int i = blockIdx.x * blockDim.x + threadIdx.x;
  if (i < n) p[i] = 0.f;
}

// ---------------------------------------------------------------------------
// Kernel 5: Wo branch   s[b,r,w] = 2*detal * sum_v Xf[r,v]*Tw[v,w]
// grid (128 row-tiles, 8 batches), block 256 (8 waves, 4 tiles each)
// ---------------------------------------------------------------------------
__global__ __launch_bounds__(256) void k_wo(const float* __restrict__ inp,
                                            const float* __restrict__ Tw,
                                            const float* __restrict__ detal,
                                            float* __restrict__ s) {
  __shared__ float as[64][68];    // Xf rows [row][k-chunk]
  __shared__ float bs[64][132];   // Tw [k-chunk][w]
  const int b   = blockIdx.y;
  const int r0  = blockIdx.x * 64;
  const int tid = threadIdx.x, lane = tid & 31, wv = tid >> 5;
  const int hf  = lane >> 4, l16 = lane & 15;
  const int rb  = wv >> 1, cb0 = (wv & 1) * 4;
  v8f acc[4];
#pragma unroll
  for (int t = 0; t < 4; ++t) acc[t] = vzero8();

  for (int k0 = 0; k0 < 128; k0 += 64) {
#pragma unroll
    for (int i = tid; i < 64 * 16; i += 256) {
      int rr = i >> 4, k4 = (i & 15) << 2;
      async_ld_b128(&as[rr][k4],
                    inp + ((size_t)(b * 8192 + r0 + rr)) * HWDIM + k0 + k4);
    }
#pragma unroll
    for (int i = tid; i < 64 * 32; i += 256) {
      int kk = i >> 5, n4 = (i & 31) << 2;
      async_ld_b128(&bs[kk][n4],
                    Tw + ((size_t)(b * 128 + k0 + kk)) * 128 + n4);
    }
    async_wait0();
    __syncthreads();
#pragma unroll
    for (int kk = 0; kk < 64; kk += 4) {
      v2f a;
      a.x = as[rb * 16 + l16][kk + 2 * hf];
      a.y = as[rb * 16 + l16][kk + 2 * hf + 1];
#pragma unroll
      for (int t = 0; t < 4; ++t) {
        const int nb = (cb0 + t) * 16;
        v2f bb;
        bb.x = bs[kk + 2 * hf][nb + l16];
        bb.y = bs[kk + 2 * hf + 1][nb + l16];
        acc[t] = wmma4(a, bb, acc[t]);
      }
    }
    __syncthreads();
  }
  const float dt2 = 2.0f * detal[0];
#pragma unroll
  for (int v = 0; v < 8; ++v) {
    const int r = r0 + rb * 16 + v + 8 * hf;
#pragma unroll
    for (int t = 0; t < 4; ++t)
      s[((size_t)(b * 8192 + r)) * HWDIM + (cb0 + t) * 16 + l16] = dt2 * acc[t][v];
  }
}

// ---------------------------------------------------------------------------
// Kernel 6: Co branch (raw-reshape layout): s_flat[b, n*64+d] += detal *
//           sum_e inp[b,e,pixel n]*Tc[e,d]
// grid (128 pixel-tiles, 8 batches), block 256 (8 waves; wave = 16 pixels x 64 d)
// ---------------------------------------------------------------------------
__global__ __launch_bounds__(256) void k_co(const float* __restrict__ inp,
                                            const float* __restrict__ Tc,
                                            const float* __restrict__ detal,
                                            float* __restrict__ s) {
  __shared__ float es[64][132];   // [channel e][pixel]
  __shared__ float ts[64][68];    // Tc [e][d]
  const int b   = blockIdx.y;
  const int n0  = blockIdx.x * 128;
  const int tid = threadIdx.x, lane = tid & 31, wv = tid >> 5;
  const int hf  = lane >> 4, l16 = lane & 15;
  const int rb  = wv;             // 8 row-blocks of 16 pixels
  v8f acc[4];
#pragma unroll
  for (int t = 0; t < 4; ++t) acc[t] = vzero8();

#pragma unroll
  for (int i = tid; i < 64 * 32; i += 256) {
    int e = i >> 5, n4 = (i & 31) << 2;
    async_ld_b128(&es[e][n4], inp + ((size_t)(b * C_N + e)) * HW + n0 + n4);
  }
#pragma unroll
  for (int i = tid; i < 64 * 16; i += 256) {
    int e = i >> 4, d4 = (i & 15) << 2;
    async_ld_b128(&ts[e][d4], Tc + ((size_t)(b * C_N + e)) * C_N + d4);
  }
  async_wait0();
  __syncthreads();
#pragma unroll
  for (int kk = 0; kk < 64; kk += 4) {   // kk indexes channels e (the K dim)
    v2f a;
    a.x = es[kk + 2 * hf][rb * 16 + l16];
    a.y = es[kk + 2 * hf + 1][rb * 16 + l16];
#pragma unroll
    for (int t = 0; t < 4; ++t) {
      v2f bb;
      bb.x = ts[kk + 2 * hf][t * 16 + l16];
      bb.y = ts[kk + 2 * hf + 1][t * 16 + l16];
      acc[t] = wmma4(a, bb, acc[t]);
    }
  }
  const float dt = detal[0];
#pragma unroll
  for (int v = 0; v < 8; ++v) {
    const int n = n0 + rb * 16 + v + 8 * hf;
#pragma unroll
    for (int t = 0; t < 4; ++t) {
      size_t f = (size_t)b * (HW * (size_t)C_N) + (size_t)n * C_N + t * 16 + l16;
      s[f] += dt * acc[t][v];
    }
  }
}

// ---------------------------------------------------------------------------
// Kernel 7: OutConv + bias + residual
//   out[b,o,p] = sum_c w_out[o,c]*s[b,c,p] + b_out[o] + x[b,o,p]
// grid (128 pixel-tiles, 4 o-groups, 8 batches), block 256 (8 waves, 4 tiles)
// ---------------------------------------------------------------------------
__global__ __launch_bounds__(256) void k_outconv(const float* __restrict__ s,
                                                 const float* __restrict__ w_out,
                                                 const float* __restrict__ b_out,
                                                 const float* __restrict__ x,
                                                 float* __restrict__ out) {
  __shared__ float ss[64][132];   // s [c][pixel]
  __shared__ float wsm[64][68];   // w_out [o][c]
  const int b   = blockIdx.z;
  const int o0  = blockIdx.y * 64;
  const int p0  = blockIdx.x * 128;
  const int tid = threadIdx.x, lane = tid & 31, wv = tid >> 5;
  const int hf  = lane >> 4, l16 = lane & 15;
  const int rb  = wv >> 1, cb0 = (wv & 1) * 4;
  v8f acc[4];
#pragma unroll
  for (int t = 0; t < 4; ++t) acc[t] = vzero8();

#pragma unroll
  for (int i = tid; i < 64 * 32; i += 256) {
    int c = i >> 5, n4 = (i & 31) << 2;
    async_ld_b128(&ss[c][n4], s + ((size_t)(b * C_N + c)) * HW + p0 + n4);
  }
#pragma unroll
  for (int i = tid; i < 64 * 16; i += 256) {
    int m = i >> 4, c4 = (i & 15) << 2;
    async_ld_b128(&wsm[m][c4], w_out + (size_t)(o0 + m) * C_N + c4);
  }
  async_wait0();
  __syncthreads();
#pragma unroll
  for (int kk = 0; kk < 64; kk += 4) {
    v2f a;
    a.x = wsm[rb * 16 + l16][kk + 2 * hf];
    a.y = wsm[rb * 16 + l16][kk + 2 * hf + 1];
#pragma unroll
    for (int t = 0; t < 4; ++t) {
      const int nb = (cb0 + t) * 16;
      v2f bb;
      bb.x = ss[kk + 2 * hf][nb + l16];
      bb.y = ss[kk + 2 * hf + 1][nb + l16];
      acc[t] = wmma4(a, bb, acc[t]);
    }
  }
#pragma unroll
  for (int v = 0; v < 8; ++v) {
    const int o = o0 + rb * 16 + v + 8 * hf;
    const float bo = b_out[o];
#pragma unroll
    for (int t = 0; t < 4; ++t) {
      size_t idx = ((size_t)(b * ICH + o)) * HW + p0 + (cb0 + t) * 16 + l16;
      out[idx] = acc[t][v] + bo + x[idx];
    }
  }
}

// ---------------------------------------------------------------------------
extern "C" void kernel_launch(void* const* d_in, const int* in_sizes, int n_in,
                              void* d_out, int out_size, void* d_ws, size_t ws_size,
                              hipStream_t stream) {
  (void)in_sizes; (void)n_in; (void)out_size; (void)ws_size;
  const float* x     = (const float*)d_in[0];
  const float* w_in  = (const float*)d_in[1];
  const float* b_in  = (const float*)d_in[2];
  const float* w_out = (const float*)d_in[3];
  const float* b_out = (const float*)d_in[4];
  const float* detal = (const float*)d_in[5];
  float* out = (float*)d_out;
  float* ws  = (float*)d_ws;

  // Workspace layout (floats); requires ws_size >= 68,419,584 bytes.
  float* inp = ws;                   // [8][64][16384]
  float* s   = ws + 8388608;         // [8][64][16384]
  float* Gw  = ws + 16777216;        // [8][128][128]
  float* Gc  = ws + 16908288;        // [8][64][64]
  float* Tw  = ws + 16941056;        // [8][128][128]
  float* Tc  = ws + 17072128;        // [8][64][64]

  k_zero<<<dim3(640), 256, 0, stream>>>(Gw, 163840);                 // Gw + Gc
  k_inconv<<<dim3(128, 8), 256, 0, stream>>>(x, w_in, b_in, inp);
  k_gramw<<<dim3(16, 8), 1024, 0, stream>>>(inp, Gw);
  k_gramc<<<dim3(16, 8), 512, 0, stream>>>(inp, Gc);
  k_colsoftmax<<<dim3(8), 128, 0, stream>>>(Gw, Tw, 128);
  k_colsoftmax<<<dim3(8), 64, 0, stream>>>(Gc, Tc, 64);
  k_wo<<<dim3(128, 8), 256, 0, stream>>>(inp, Tw, detal, s);
  k_co<<<dim3(128, 8), 256, 0, stream>>>(inp, Tc, detal, s);
  k_outconv<<<dim3(128, 4, 8), 256, 0, stream>>>(s, w_out, b_out, x, out);
}